// GCN_9105330667739
// MI455X (gfx1250) — compile-verified
//
#include <hip/hip_runtime.h>

// ---------------------------------------------------------------------------
// GCN (2x GCNConv + mean-pool + MLP) for MI455X / gfx1250, wave32.
//
// Roofline: both GCN GEMMs are bandwidth-bound (3.28 GFLOP vs ~100MB traffic
// at 23.3 TB/s), so fp32 WMMA (V_WMMA_F32_16X16X4_F32) is the right precision:
// exact reference numerics, ample matrix throughput. The edge scatter/gather
// (~3.3 GB/layer) stays inside the 192MB L2 since h + agg = 102MB.
// ---------------------------------------------------------------------------

typedef __attribute__((ext_vector_type(2))) float v2f;
typedef __attribute__((ext_vector_type(8))) float v8f;

#define D 128  // channel count of both GCN layers (== D_IN == CH[0] == CH[1])

// ---------------------------------------------------------------------------
__global__ __launch_bounds__(256) void fill_f32(float* __restrict__ p, float v, long n) {
  long i = (long)blockIdx.x * 256 + threadIdx.x;
  if (i < n) p[i] = v;
}

__global__ __launch_bounds__(256) void degree_kernel(const int* __restrict__ dst,
                                                     float* __restrict__ deg, int E) {
  int e = blockIdx.x * 256 + threadIdx.x;
  if (e < E) atomicAdd(&deg[dst[e]], 1.0f);
}

__global__ __launch_bounds__(256) void rsqrt_inplace(float* __restrict__ p, int n) {
  int i = blockIdx.x * 256 + threadIdx.x;
  if (i < n) p[i] = rsqrtf(p[i]);  // deg >= 1 always (self loop)
}

// ---------------------------------------------------------------------------
// H[M,128] = X[M,128] @ W[128,128] using V_WMMA_F32_16X16X4_F32.
// Block = 256 threads = 8 waves; block computes a 16(M) x 128(N) stripe,
// wave w owns the 16x16 tile at columns [16w, 16w+16).
//
// Wave strategy: the B stripe (128x16 of W) is loop-invariant per wave, so all
// 32 B fragments (64 VGPRs) are preloaded in one clause burst; the fully
// unrolled K loop is then pure ds_load_2addr_b64 + v_wmma (32 WMMAs).
//
// Fragment layouts (ISA 7.12.2, fp32):
//   A 16x4 : lanes 0-15 -> M=lane, VGPR0=K0,VGPR1=K0+1; lanes 16-31 -> K0+2/K0+3
//   B 4x16 : lanes 0-15 -> N=lane, rows K0/K0+1;        lanes 16-31 -> rows K0+2/K0+3
//   C 16x16: VGPR r: lanes 0-15 -> M=r, N=lane; lanes 16-31 -> M=r+8, N=lane-16
// ---------------------------------------------------------------------------
__global__ __launch_bounds__(256) void gemm_wmma_f32(const float* __restrict__ X,
                                                     const float* __restrict__ W,
                                                     float* __restrict__ H, int M) {
  __shared__ float sA[16 * D];  // 8 KB A-slab
  const int tid  = threadIdx.x;
  const int wave = tid >> 5;
  const int lane = tid & 31;
  const int m0   = blockIdx.x << 4;
  const bool fullTile = (m0 + 16 <= M);

  const int n0   = wave << 4;
  const int half = lane >> 4;   // 0: lanes 0-15, 1: lanes 16-31
  const int l15  = lane & 15;

  // Preload all B fragments for this wave's column stripe (loop-invariant).
  v2f bf[32];
#pragma unroll
  for (int k = 0; k < 32; ++k) {
    const int ka = 4 * k + half * 2;
    bf[k].x = W[(size_t)ka * D + n0 + l15];
    bf[k].y = W[(size_t)(ka + 1) * D + n0 + l15];
  }

  // Stage the 16x128 A-slab in LDS (b128 path on the common full tile).
  if (fullTile) {
    const float4* Xv = (const float4*)(X + (size_t)m0 * D);
    float4* sAv = (float4*)sA;
    for (int i = tid; i < 16 * D / 4; i += 256) sAv[i] = Xv[i];
    // Prefetch next block's A-slab (global_prefetch_b8) while we compute.
    if (tid < 16 && m0 + 16 + tid < M)
      __builtin_prefetch(X + (size_t)(m0 + 16 + tid) * D, 0, 1);
  } else {
    for (int i = tid; i < 16 * D; i += 256) {
      int r = i >> 7, c = i & (D - 1);
      int gr = m0 + r;
      sA[i] = (gr < M) ? X[(size_t)gr * D + c] : 0.0f;
    }
  }
  __syncthreads();

  v8f acc = {};
#pragma unroll
  for (int k = 0; k < 32; ++k) {
    const int ka = 4 * k + half * 2;
    v2f a;
    a.x = sA[l15 * D + ka];
    a.y = sA[l15 * D + ka + 1];
    // (neg_a, A, neg_b, B, c_mod, C, reuse_a, reuse_b)
    acc = __builtin_amdgcn_wmma_f32_16x16x4_f32(false, a, false, bf[k], (short)0,
                                                acc, false, false);
  }

  const int mBase = m0 + half * 8;
  if (fullTile) {
#pragma unroll
    for (int r = 0; r < 8; ++r)
      H[(size_t)(mBase + r) * D + n0 + l15] = acc[r];
  } else {
#pragma unroll
    for (int r = 0; r < 8; ++r)
      if (mBase + r < M) H[(size_t)(mBase + r) * D + n0 + l15] = acc[r];
  }
}

// ---------------------------------------------------------------------------
// One wave32 per edge: each lane owns a float4 of the 128-channel row.
// agg[dst] += h[src] * (dinv[src]*dinv[dst])   via global_atomic_add_f32.
// h and agg both live in the 192MB L2, so this runs at L2 rate.
// ---------------------------------------------------------------------------
__global__ __launch_bounds__(256) void aggregate_edges(const float* __restrict__ H,
                                                       const int* __restrict__ src,
                                                       const int* __restrict__ dst,
                                                       const float* __restrict__ dinv,
                                                       float* __restrict__ AGG, int E) {
  const int gw   = (int)(((long)blockIdx.x * 256 + threadIdx.x) >> 5);
  const int lane = threadIdx.x & 31;
  if (gw >= E) return;
  const int s = __builtin_amdgcn_readfirstlane(src[gw]);
  const int d = __builtin_amdgcn_readfirstlane(dst[gw]);
  const float norm = dinv[s] * dinv[d];
  const float4 v = ((const float4*)(H + (size_t)s * D))[lane];
  float* out = AGG + (size_t)d * D + lane * 4;
  atomicAdd(out + 0, v.x * norm);
  atomicAdd(out + 1, v.y * norm);
  atomicAdd(out + 2, v.z * norm);
  atomicAdd(out + 3, v.w * norm);
}

// AGG = relu(AGG + H * dinv^2 + b)   (in place)
__global__ __launch_bounds__(256) void selfloop_bias_relu(float* __restrict__ AGG,
                                                          const float* __restrict__ H,
                                                          const float* __restrict__ dinv,
                                                          const float* __restrict__ b,
                                                          long n) {
  long i = (long)blockIdx.x * 256 + threadIdx.x;
  if (i >= n) return;
  int node = (int)(i >> 7);
  int c    = (int)(i & (D - 1));
  float di = dinv[node];
  float v  = AGG[i] + H[i] * di * di + b[c];
  AGG[i]   = v > 0.0f ? v : 0.0f;
}

// ---------------------------------------------------------------------------
__global__ __launch_bounds__(256) void pool_accum(const float* __restrict__ H,
                                                  const int* __restrict__ batch,
                                                  float* __restrict__ PS,
                                                  float* __restrict__ CNT, long n) {
  long i = (long)blockIdx.x * 256 + threadIdx.x;
  if (i >= n) return;
  int node = (int)(i >> 7);
  int c    = (int)(i & (D - 1));
  int g    = batch[node];
  atomicAdd(&PS[(size_t)g * D + c], H[i]);
  if (c == 0) atomicAdd(&CNT[g], 1.0f);
}

__global__ __launch_bounds__(256) void pool_div(const float* __restrict__ PS,
                                                const float* __restrict__ CNT,
                                                float* __restrict__ G, int nG) {
  int i = blockIdx.x * 256 + threadIdx.x;
  if (i >= nG * D) return;
  int g = i >> 7;
  G[i] = PS[i] / fmaxf(CNT[g], 1.0f);
}

// OUT[rows,N] = act(IN[rows,K] @ W[K,N] + B)   (tiny: 256 rows)
__global__ __launch_bounds__(256) void mlp_layer(const float* __restrict__ IN,
                                                 const float* __restrict__ W,
                                                 const float* __restrict__ B,
                                                 float* __restrict__ OUT, int rows,
                                                 int K, int N, int doRelu) {
  int i = blockIdx.x * 256 + threadIdx.x;
  if (i >= rows * N) return;
  int r = i / N, c = i % N;
  float acc = B[c];
  for (int k = 0; k < K; ++k)
    acc = fmaf(IN[(size_t)r * K + k], W[(size_t)k * N + c], acc);
  OUT[i] = doRelu ? fmaxf(acc, 0.0f) : acc;
}

// ---------------------------------------------------------------------------
extern "C" void kernel_launch(void* const* d_in, const int* in_sizes, int n_in,
                              void* d_out, int out_size, void* d_ws, size_t ws_size,
                              hipStream_t stream) {
  const float* x     = (const float*)d_in[0];
  const int*   ei    = (const int*)d_in[1];   // [2, E]
  const int*   batch = (const int*)d_in[2];
  const float* W1    = (const float*)d_in[3];
  const float* b1    = (const float*)d_in[4];
  const float* W2    = (const float*)d_in[5];
  const float* b2    = (const float*)d_in[6];
  const float* Wm1   = (const float*)d_in[7];
  const float* bm1   = (const float*)d_in[8];
  const float* Wm2   = (const float*)d_in[9];
  const float* bm2   = (const float*)d_in[10];
  const float* Wm3   = (const float*)d_in[11];
  const float* bm3   = (const float*)d_in[12];
  (void)n_in; (void)ws_size;

  const int N = in_sizes[0] / D;  // 100000 nodes
  const int E = in_sizes[1] / 2;  // 1600000 edges
  const int G = out_size;         // 256 graphs (output is [G,1] fp32)
  const int* src = ei;
  const int* dst = ei + E;

  // workspace carve-up (floats); ~104 MB total
  float* ws  = (float*)d_ws;
  size_t off = 0;
  float* deg = ws + off; off += (size_t)N;           // deg, then dinv in place
  off = (off + 3) & ~(size_t)3;                      // 16B align for float4 loads
  float* Hb  = ws + off; off += (size_t)N * D;       // GEMM output
  float* AGG = ws + off; off += (size_t)N * D;       // aggregation / layer output
  float* PS  = ws + off; off += (size_t)G * D;       // pool sums
  float* CNT = ws + off; off += (size_t)G;           // pool counts
  float* Gp  = ws + off; off += (size_t)G * D;       // pooled features
  float* G1  = ws + off; off += (size_t)G * 500;
  float* G2  = ws + off; off += (size_t)G * 100;

  const long nF = (long)N * D;
  dim3 blk(256);
  auto cdiv = [](long a, long b) { return (unsigned)((a + b - 1) / b); };

  // symmetric-norm degrees (deg = 1 + indegree; dinv = rsqrt)
  fill_f32<<<cdiv(N, 256), blk, 0, stream>>>(deg, 1.0f, N);
  degree_kernel<<<cdiv(E, 256), blk, 0, stream>>>(dst, deg, E);
  rsqrt_inplace<<<cdiv(N, 256), blk, 0, stream>>>(deg, N);

  // ---- GCN layer 1 ----
  gemm_wmma_f32<<<cdiv(N, 16), blk, 0, stream>>>(x, W1, Hb, N);
  fill_f32<<<cdiv(nF, 256), blk, 0, stream>>>(AGG, 0.0f, nF);
  aggregate_edges<<<cdiv((long)E * 32, 256), blk, 0, stream>>>(Hb, src, dst, deg, AGG, E);
  selfloop_bias_relu<<<cdiv(nF, 256), blk, 0, stream>>>(AGG, Hb, deg, b1, nF);

  // ---- GCN layer 2 (input = AGG, consumed by the GEMM before re-zeroing) ----
  gemm_wmma_f32<<<cdiv(N, 16), blk, 0, stream>>>(AGG, W2, Hb, N);
  fill_f32<<<cdiv(nF, 256), blk, 0, stream>>>(AGG, 0.0f, nF);
  aggregate_edges<<<cdiv((long)E * 32, 256), blk, 0, stream>>>(Hb, src, dst, deg, AGG, E);
  selfloop_bias_relu<<<cdiv(nF, 256), blk, 0, stream>>>(AGG, Hb, deg, b2, nF);

  // ---- global mean pool ----
  fill_f32<<<cdiv((long)G * D, 256), blk, 0, stream>>>(PS, 0.0f, (long)G * D);
  fill_f32<<<1, blk, 0, stream>>>(CNT, 0.0f, G);
  pool_accum<<<cdiv(nF, 256), blk, 0, stream>>>(AGG, batch, PS, CNT, nF);
  pool_div<<<cdiv((long)G * D, 256), blk, 0, stream>>>(PS, CNT, Gp, G);

  // ---- MLP head ----
  mlp_layer<<<cdiv((long)G * 500, 256), blk, 0, stream>>>(Gp, Wm1, bm1, G1, G, D, 500, 1);
  mlp_layer<<<cdiv((long)G * 100, 256), blk, 0, stream>>>(G1, Wm2, bm2, G2, G, 500, 100, 1);
  mlp_layer<<<cdiv(G, 256), blk, 0, stream>>>(G2, Wm3, bm3, (float*)d_out, G, 100, 1, 0);
}